// GraphConvolution_57071525430034
// MI455X (gfx1250) — compile-verified
//
#include <hip/hip_runtime.h>
#include <hip/hip_bf16.h>
#include <math.h>

#define N_NODES   100000
#define N_EDGES   1600000
#define C         128
#define E_TOTAL   (N_EDGES + N_NODES)
#define EPS_BN    1e-5f
#define NEG_SLOPE 0.2f

typedef __attribute__((ext_vector_type(2))) float v2f;
typedef __attribute__((ext_vector_type(8))) float v8f;
typedef unsigned int u32x4 __attribute__((ext_vector_type(4)));
typedef int          i32x4 __attribute__((ext_vector_type(4)));
typedef int          i32x8 __attribute__((ext_vector_type(8)));

// LDS row stride for W chunks: 128 data dwords + 8 pad dwords.
//  - 2*136 % 64 == 16 -> the two lane halves (rows k, k+2) of one ds_load
//    hit disjoint 16-bank groups.
//  - TDM reproduces this layout natively: pad_interval=6 (every 128 dwords),
//    pad_amount=7 (8 dwords).
#define SWSTRIDE   136
#define CHUNK_ROWS 32
#define N_CHUNKS   4

// -------------------------------------------------------------------------
// TDM: DMA one 32x128 f32 tile of W from global into LDS with row padding.
// D# packing per CDNA5 ISA ch.8 (2D tile -> groups 2/3 zero).
// -------------------------------------------------------------------------
__device__ __forceinline__ void tdm_load_w_chunk(const float* __restrict__ gsrc,
                                                 unsigned lds_byte_off) {
  const unsigned long long ga = (unsigned long long)(size_t)gsrc;
  u32x4 g0;
  g0.x = 1u;                                            // count=1, user D#
  g0.y = lds_byte_off;                                  // lds_addr (bytes)
  g0.z = (unsigned)(ga & 0xFFFFFFFFu);                  // global_addr[31:0]
  g0.w = (unsigned)((ga >> 32) & 0x01FFFFFFu)           // global_addr[56:32]
       | (2u << 30);                                    // type=2 ("image")
  i32x8 g1;
  g1[0] = (int)((2u << 16)        // data_size = 2 -> 4-byte elements
              | (1u << 20)        // pad_enable
              | (6u << 22)        // pad_interval: every 128 dwords
              | (7u << 25));      // pad_amount:  8 dwords
  g1[1] = (int)(128u << 16);      // tensor_dim0 = 128 (bits 79:48, low half)
  g1[2] = (int)(32u  << 16);      // tensor_dim1 = 32  (bits 111:80, low half)
  g1[3] = (int)(128u << 16);      // tile_dim0   = 128 (bits 127:112)
  g1[4] = 32;                     // tile_dim1   = 32  (bits 143:128)
  g1[5] = 128;                    // tensor_dim0_stride = 128 (bits 207:160)
  g1[6] = 0;
  g1[7] = 0;
  const i32x4 z4 = {};
#if defined(__clang_major__) && __clang_major__ >= 23
  const i32x8 z8 = {};
  __builtin_amdgcn_tensor_load_to_lds(g0, g1, z4, z4, z8, 0);
#else
  __builtin_amdgcn_tensor_load_to_lds(g0, g1, z4, z4, 0);
#endif
}

// -------------------------------------------------------------------------
// h = x @ W  via V_WMMA_F32_16X16X4_F32.
// Block: 256 thr = 8 waves; each wave owns 16 rows x all 128 cols (8 tiles).
// W staged through LDS by the Tensor Data Mover in four 32-row chunks,
// double-buffered: wave 0 issues DMAs, waits on TENSORcnt, workgroup barrier
// publishes each chunk.
// -------------------------------------------------------------------------
__global__ __launch_bounds__(256) void gat_gemm_wmma(const float* __restrict__ x,
                                                     const float* __restrict__ W,
                                                     float* __restrict__ h, int n) {
  __shared__ float sW[2][CHUNK_ROWS * SWSTRIDE];        // 2 x 17,408 B
  const int tid   = threadIdx.x;
  const int wave  = tid >> 5;
  const int lane  = tid & 31;
  const int l15   = lane & 15;
  const int khalf = (lane >> 4) << 1;          // 0 or 2 (K-split across lane halves)
  const int rowBase = blockIdx.x * 128 + wave * 16;

  int rowA = rowBase + l15;                    // A-fragment source row (clamped)
  if (rowA > n - 1) rowA = n - 1;
  const float* xrow = x + (long)rowA * C;

  v8f acc[8];
  v8f zero = {};
#pragma unroll
  for (int t = 0; t < 8; ++t) acc[t] = zero;

  // prime the pipeline: chunks 0 and 1 in flight
  if (wave == 0) {
    tdm_load_w_chunk(W,                  (unsigned)(size_t)&sW[0][0]);
    tdm_load_w_chunk(W + CHUNK_ROWS * C, (unsigned)(size_t)&sW[1][0]);
  }

#pragma unroll
  for (int chunk = 0; chunk < N_CHUNKS; ++chunk) {
    if (wave == 0) {
      if (chunk == N_CHUNKS - 1) __builtin_amdgcn_s_wait_tensorcnt(0);
      else                       __builtin_amdgcn_s_wait_tensorcnt(1);
    }
    __syncthreads();                           // publish chunk to all waves

    const float* buf = &sW[chunk & 1][0];
#pragma unroll
    for (int k4 = 0; k4 < CHUNK_ROWS; k4 += 4) {
      // A frag: (M=l15, K=base+k4+khalf), (M, K+1) -> one b64 load
      const v2f a = *(const v2f*)(xrow + chunk * CHUNK_ROWS + k4 + khalf);
#pragma unroll
      for (int t = 0; t < 8; ++t) {
        const int col = t * 16 + l15;
        v2f b;
        b.x = buf[(k4 + khalf) * SWSTRIDE + col];
        b.y = buf[(k4 + khalf + 1) * SWSTRIDE + col];
        acc[t] = __builtin_amdgcn_wmma_f32_16x16x4_f32(
            false, a, false, b, (short)0, acc[t], false, false);
      }
    }
    __syncthreads();                           // all waves done with this buffer

    if (wave == 0 && chunk + 2 < N_CHUNKS)     // refill the buffer just freed
      tdm_load_w_chunk(W + (chunk + 2) * CHUNK_ROWS * C,
                       (unsigned)(size_t)&sW[chunk & 1][0]);
  }

  // C layout: VGPR r, lane l -> (M = r + 8*(l>=16), N = l%16)
  const int rOff = (lane >> 4) * 8;
#pragma unroll
  for (int t = 0; t < 8; ++t) {
    const int col = t * 16 + l15;
#pragma unroll
    for (int r = 0; r < 8; ++r) {
      const int row = rowBase + rOff + r;
      if (row < n) h[(long)row * C + col] = acc[t][r];
    }
  }
}

// -------------------------------------------------------------------------
// a_s[i] = h[i,:].att_src ; a_d[i] = h[i,:].att_dst   (one wave per node)
// -------------------------------------------------------------------------
__global__ __launch_bounds__(256) void gat_att(const float* __restrict__ h,
                                               const float* __restrict__ att_s,
                                               const float* __restrict__ att_d,
                                               float* __restrict__ a_s,
                                               float* __restrict__ a_d, int n) {
  const int wave = threadIdx.x >> 5, lane = threadIdx.x & 31;
  const int node = blockIdx.x * 8 + wave;
  if (node >= n) return;
  const float* hr = h + (long)node * C;
  float ss = 0.f, dd = 0.f;
#pragma unroll
  for (int j = 0; j < 4; ++j) {
    const float v = hr[j * 32 + lane];
    ss += v * att_s[j * 32 + lane];
    dd += v * att_d[j * 32 + lane];
  }
#pragma unroll
  for (int off = 16; off > 0; off >>= 1) {
    ss += __shfl_down(ss, off, 32);
    dd += __shfl_down(dd, off, 32);
  }
  if (lane == 0) { a_s[node] = ss; a_d[node] = dd; }
}

// -------------------------------------------------------------------------
// Edge phase
// -------------------------------------------------------------------------
__device__ __forceinline__ void edge_pair(const int* __restrict__ ei, int id,
                                          int& s, int& d) {
  if (id < N_EDGES) { s = ei[id]; d = ei[N_EDGES + id]; }
  else              { s = d = id - N_EDGES; }           // self loops
}

__global__ __launch_bounds__(256) void edge_score(const int* __restrict__ ei,
                                                  const float* __restrict__ a_s,
                                                  const float* __restrict__ a_d,
                                                  float* __restrict__ e,
                                                  float* __restrict__ m) {
  const int id = blockIdx.x * 256 + threadIdx.x;
  if (id >= E_TOTAL) return;
  int s, d; edge_pair(ei, id, s, d);
  float v = a_s[s] + a_d[d];
  v = v > 0.f ? v : v * NEG_SLOPE;
  e[id] = v;
  // float atomic max via int punning (m initialized to -inf)
  if (v >= 0.f) atomicMax((int*)m + d, __float_as_int(v));
  else          atomicMin((unsigned int*)m + d, __float_as_uint(v));
}

__global__ __launch_bounds__(256) void edge_exp(const int* __restrict__ ei,
                                                float* __restrict__ e,
                                                const float* __restrict__ m,
                                                float* __restrict__ denom) {
  const int id = blockIdx.x * 256 + threadIdx.x;
  if (id >= E_TOTAL) return;
  int s, d; edge_pair(ei, id, s, d);
  const float ex = expf(e[id] - m[d]);
  e[id] = ex;
  atomicAdd(&denom[d], ex);
}

// one wave per edge: gather h[src] (float4/lane), scatter-add alpha*h into agg[dst]
__global__ __launch_bounds__(256) void edge_agg(const int* __restrict__ ei,
                                                const float* __restrict__ ex,
                                                const float* __restrict__ denom,
                                                const float* __restrict__ h,
                                                float* __restrict__ agg,
                                                float* __restrict__ alpha_out) {
  const int wave = threadIdx.x >> 5, lane = threadIdx.x & 31;
  const int id = blockIdx.x * 8 + wave;
  if (id >= E_TOTAL) return;
  int s, d; edge_pair(ei, id, s, d);
  const float alpha = ex[id] / (denom[d] + 1e-16f);
  if (lane == 0) alpha_out[id] = alpha;
  const float4 hv = ((const float4*)(h + (long)s * C))[lane];
  float* ag = agg + (long)d * C + lane * 4;
  atomicAdd(ag + 0, alpha * hv.x);
  atomicAdd(ag + 1, alpha * hv.y);
  atomicAdd(ag + 2, alpha * hv.z);
  atomicAdd(ag + 3, alpha * hv.w);
}

// -------------------------------------------------------------------------
// BatchNorm over axis 0 (biased var), fused with bias-add + ReLU
// -------------------------------------------------------------------------
__global__ __launch_bounds__(128) void bn_stats(const float* __restrict__ agg,
                                                const float* __restrict__ bias,
                                                float* __restrict__ colsum,
                                                float* __restrict__ colsq, int n) {
  const int c = threadIdx.x;
  const float bc = bias[c];
  const int r0 = blockIdx.x * 256;
  int r1 = r0 + 256; if (r1 > n) r1 = n;
  float s = 0.f, s2 = 0.f;
  for (int r = r0; r < r1; ++r) {
    float v = agg[(long)r * C + c] + bc;
    v = v > 0.f ? v : 0.f;
    s += v; s2 += v * v;
  }
  atomicAdd(&colsum[c], s);
  atomicAdd(&colsq[c], s2);
}

__global__ __launch_bounds__(256) void bn_apply(const float* __restrict__ agg,
                                                const float* __restrict__ bias,
                                                const float* __restrict__ colsum,
                                                const float* __restrict__ colsq,
                                                const float* __restrict__ gamma,
                                                const float* __restrict__ beta,
                                                float* __restrict__ x_next,
                                                float* __restrict__ out_x, int n) {
  const long i = (long)blockIdx.x * 256 + threadIdx.x;
  if (i >= (long)n * C) return;
  const int c   = (int)(i & (C - 1));
  const int row = (int)(i >> 7);
  float v = agg[i] + bias[c];
  v = v > 0.f ? v : 0.f;
  const float invN = 1.0f / (float)n;
  const float mu   = colsum[c] * invN;
  const float var  = colsq[c] * invN - mu * mu;
  const float y = (v - mu) * rsqrtf(var + EPS_BN) * gamma[c] + beta[c];
  x_next[i] = y;
  out_x[(long)row * 384 + c] = y;     // jump-concat slice
}

__global__ void fill_f32(float* __restrict__ p, float v, long n) {
  const long i = (long)blockIdx.x * 256 + threadIdx.x;
  if (i < n) p[i] = v;
}

// -------------------------------------------------------------------------
static inline int gblocks(long n) { return (int)((n + 255) / 256); }

extern "C" void kernel_launch(void* const* d_in, const int* in_sizes, int n_in,
                              void* d_out, int out_size, void* d_ws, size_t ws_size,
                              hipStream_t stream) {
  const float* x  = (const float*)d_in[0];
  const int*   ei = (const int*)d_in[1];
  const float* W[3]  = {(const float*)d_in[2],  (const float*)d_in[8],  (const float*)d_in[14]};
  const float* as[3] = {(const float*)d_in[3],  (const float*)d_in[9],  (const float*)d_in[15]};
  const float* ad[3] = {(const float*)d_in[4],  (const float*)d_in[10], (const float*)d_in[16]};
  const float* bb[3] = {(const float*)d_in[5],  (const float*)d_in[11], (const float*)d_in[17]};
  const float* gm[3] = {(const float*)d_in[6],  (const float*)d_in[12], (const float*)d_in[18]};
  const float* bt[3] = {(const float*)d_in[7],  (const float*)d_in[13], (const float*)d_in[19]};

  float* ws = (float*)d_ws;
  size_t off = 0;
  float* h      = ws + off; off += (size_t)N_NODES * C;   // 51.2 MB (L2-resident)
  float* agg    = ws + off; off += (size_t)N_NODES * C;   // 51.2 MB (L2-resident)
  float* xbuf   = ws + off; off += (size_t)N_NODES * C;   // next-layer input
  float* e_buf  = ws + off; off += (size_t)E_TOTAL;
  float* a_s    = ws + off; off += N_NODES;
  float* a_dv   = ws + off; off += N_NODES;
  float* mbuf   = ws + off; off += N_NODES;
  float* denom  = ws + off; off += N_NODES;
  float* colsum = ws + off; off += C;
  float* colsq  = ws + off; off += C;

  float* out = (float*)d_out;
  float* alpha0 = out + (size_t)N_NODES * 384;            // a1,a2,a3 follow x-concat

  const float* x_in = x;
  for (int L = 0; L < 3; ++L) {
    fill_f32<<<gblocks(N_NODES), 256, 0, stream>>>(mbuf, -INFINITY, N_NODES);
    fill_f32<<<gblocks(N_NODES), 256, 0, stream>>>(denom, 0.f, N_NODES);
    fill_f32<<<gblocks((long)N_NODES * C), 256, 0, stream>>>(agg, 0.f, (long)N_NODES * C);
    fill_f32<<<1, 256, 0, stream>>>(colsum, 0.f, C);
    fill_f32<<<1, 256, 0, stream>>>(colsq, 0.f, C);

    gat_gemm_wmma<<<(N_NODES + 127) / 128, 256, 0, stream>>>(x_in, W[L], h, N_NODES);
    gat_att<<<(N_NODES + 7) / 8, 256, 0, stream>>>(h, as[L], ad[L], a_s, a_dv, N_NODES);
    edge_score<<<gblocks(E_TOTAL), 256, 0, stream>>>(ei, a_s, a_dv, e_buf, mbuf);
    edge_exp<<<gblocks(E_TOTAL), 256, 0, stream>>>(ei, e_buf, mbuf, denom);
    edge_agg<<<(E_TOTAL + 7) / 8, 256, 0, stream>>>(ei, e_buf, denom, h, agg,
                                                    alpha0 + (size_t)L * E_TOTAL);
    bn_stats<<<(N_NODES + 255) / 256, 128, 0, stream>>>(agg, bb[L], colsum, colsq, N_NODES);
    bn_apply<<<gblocks((long)N_NODES * C), 256, 0, stream>>>(
        agg, bb[L], colsum, colsq, gm[L], bt[L], xbuf, out + (size_t)L * C, N_NODES);

    x_in = xbuf;   // safe: previous contents consumed by this layer's GEMM
  }
}